// QuantumRecurrentUnit_38835094291191
// MI455X (gfx1250) — compile-verified
//
#include <hip/hip_runtime.h>
#include <math.h>

// QuantumRecurrentUnit: the 10-qubit RY-embed + RY-layer + CNOT-ring circuit
// collapses analytically (product state + Heisenberg-picture CNOT pushback):
//   th_q   = x_q + w_q ; c_q = cos(th_q)
//   out[0] = c1*c2*...*c9
//   out[k] = c0*c1*...*ck          (k = 1..9)
// One thread per sample; x slice staged to LDS via gfx1250 async-DMA loads.

#define NQ 10
#define BLK 64

__global__ __launch_bounds__(BLK) void qru_analytic_kernel(
    const float* __restrict__ x,      // (nsamp, 16), first 10 used
    const float* __restrict__ w,      // (10,)
    float* __restrict__ out,          // (nsamp, 10)
    int nsamp) {
  __shared__ float tile[BLK * 12];    // 48 B per sample, 16B-aligned chunks
  const int lt = threadIdx.x;
  const int t  = blockIdx.x * BLK + lt;
  if (t >= nsamp) return;

  // --- gfx1250 async load to LDS (ASYNCcnt path) ------------------------
  // GVS addressing: global = s[base] + v[goff] + inst_offset
  //                 lds    = v[lds_base]       + inst_offset
  unsigned lds_base = (unsigned)(size_t)(&tile[lt * 12]);
  unsigned goff     = (unsigned)t * 64u;  // 16 floats * 4 B per sample in x
  asm volatile(
      "global_load_async_to_lds_b128 %0, %1, %2 offset:0\n\t"
      "global_load_async_to_lds_b128 %0, %1, %2 offset:16\n\t"
      "global_load_async_to_lds_b128 %0, %1, %2 offset:32\n\t"
      :
      : "v"(lds_base), "v"(goff), "s"(x)
      : "memory");

  // Uniform weights -> scalar (s_load) path while the DMA is in flight.
  float wv[NQ];
#pragma unroll
  for (int q = 0; q < NQ; ++q) wv[q] = w[q];

  asm volatile("s_wait_asynccnt 0x0" ::: "memory");

  // Each lane reads back only its own staged bytes (no cross-wave dep).
  float xv[12];
  *(float4*)&xv[0] = *(const float4*)&tile[lt * 12 + 0];
  *(float4*)&xv[4] = *(const float4*)&tile[lt * 12 + 4];
  *(float4*)&xv[8] = *(const float4*)&tile[lt * 12 + 8];

  float c[NQ];
#pragma unroll
  for (int q = 0; q < NQ; ++q) c[q] = cosf(xv[q] + wv[q]);

  float o[NQ];
  float p = c[0];
#pragma unroll
  for (int k = 1; k < NQ; ++k) { p *= c[k]; o[k] = p; }
  float s = c[1];
#pragma unroll
  for (int q = 2; q < NQ; ++q) s *= c[q];
  o[0] = s;

  // 40*t bytes is 8-aligned -> five b64 stores.
  float2* o2 = (float2*)(out + (size_t)t * NQ);
#pragma unroll
  for (int i = 0; i < 5; ++i) o2[i] = make_float2(o[2 * i], o[2 * i + 1]);
}

extern "C" void kernel_launch(void* const* d_in, const int* in_sizes, int n_in,
                              void* d_out, int out_size, void* d_ws, size_t ws_size,
                              hipStream_t stream) {
  const float* x = (const float*)d_in[0];   // (64,128,16) float32
  const float* w = (const float*)d_in[1];   // (10,)       float32
  float* out = (float*)d_out;               // (64,128,10) float32

  const int nsamp = in_sizes[0] / 16;       // 8192
  const int grid = (nsamp + BLK - 1) / BLK; // 128 blocks of 2 waves
  qru_analytic_kernel<<<grid, BLK, 0, stream>>>(x, w, out, nsamp);
}